// GNNPolicy_8821862826579
// MI455X (gfx1250) — compile-verified
//
#include <hip/hip_runtime.h>
#include <hip/hip_bf16.h>
#include <math.h>

typedef float v2f __attribute__((ext_vector_type(2)));
typedef float v8f __attribute__((ext_vector_type(8)));

#define HDIM 32
#define BN_EPS 1e-5f

// ---------------------------------------------------------------------------
// degree: deg[dst]++ over edges (self loops folded in later as +1)
__global__ void deg_kernel(const int* __restrict__ ei, float* __restrict__ deg, int E) {
    int e = blockIdx.x * blockDim.x + threadIdx.x;
    if (e >= E) return;
    atomicAdd(&deg[ei[E + e]], 1.0f);
}

__global__ void dinv_kernel(float* __restrict__ dinv, int n) {
    int i = blockIdx.x * blockDim.x + threadIdx.x;
    if (i >= n) return;
    dinv[i] = rsqrtf(dinv[i] + 1.0f);   // +1 self loop; deg >= 1 always
}

// ---------------------------------------------------------------------------
// Layer-1 GEMM, K=3 (too skinny for WMMA): out[N,32] = x[N,3] @ W[3,32]
__global__ void gemm_k3(const float* __restrict__ x, const float* __restrict__ W,
                        float* __restrict__ out, int n) {
    int i = blockIdx.x * blockDim.x + threadIdx.x;
    if (i >= n) return;
    float x0 = x[i * 3 + 0], x1 = x[i * 3 + 1], x2 = x[i * 3 + 2];
#pragma unroll
    for (int f = 0; f < HDIM; ++f)
        out[i * HDIM + f] = fmaf(x0, W[f], fmaf(x1, W[HDIM + f], x2 * W[2 * HDIM + f]));
}

// ---------------------------------------------------------------------------
// K=32 GEMM with V_WMMA_F32_16X16X4_F32.
// One wave32 computes one 16x16 tile of out = h[N,32] @ W[32,32].
// A (16x4 f32): lanes 0-15 -> M=0..15 {K=k,k+1}, lanes 16-31 -> {K=k+2,k+3}.
// B (4x16 f32): VGPR0 = rows K=k (lanes 0-15) / K=k+2 (lanes 16-31), VGPR1 = +1.
// D (16x16 f32): VGPR v -> row v + 8*(lane/16), col = lane%16.
__global__ void gemm32_wmma(const float* __restrict__ h, const float* __restrict__ W,
                            float* __restrict__ out, int n) {
    int wave = blockIdx.x * (blockDim.x >> 5) + (threadIdx.x >> 5);
    int lane = threadIdx.x & 31;
    int tm = wave >> 1;            // 16-row tile
    int tn = wave & 1;             // 16-col tile (32 cols -> 2 tiles)
    if (tm * 16 >= n) return;      // wave-uniform: EXEC stays all-ones
    int m   = tm * 16 + (lane & 15);
    int kg  = (lane >> 4) << 1;    // 0 or 2
    int col = tn * 16 + (lane & 15);

    v8f c = {};
#pragma unroll
    for (int ks = 0; ks < 8; ++ks) {
        int k = ks * 4;
        v2f a, b;
        a.x = h[m * HDIM + k + kg];
        a.y = h[m * HDIM + k + kg + 1];
        b.x = W[(k + kg) * HDIM + col];
        b.y = W[(k + kg + 1) * HDIM + col];
        c = __builtin_amdgcn_wmma_f32_16x16x4_f32(false, a, false, b,
                                                  (short)0, c, false, false);
    }
#pragma unroll
    for (int v = 0; v < 8; ++v) {
        int row = tm * 16 + v + ((lane >> 4) << 3);
        out[row * HDIM + tn * 16 + (lane & 15)] = c[v];
    }
}

// ---------------------------------------------------------------------------
// Edge scatter: acc[dst] += tmp[src] * dinv[src]*dinv[dst]
__global__ void scatter_edges(const int* __restrict__ ei, const float* __restrict__ tmp,
                              const float* __restrict__ dinv, float* __restrict__ acc, int E) {
    int e = blockIdx.x * blockDim.x + threadIdx.x;
    if (e >= E) return;
    int s = ei[e], d = ei[E + e];
    float w = dinv[s] * dinv[d];
    const float* ts = tmp + (size_t)s * HDIM;
    float* ad = acc + (size_t)d * HDIM;
#pragma unroll
    for (int f = 0; f < HDIM; ++f) atomicAdd(&ad[f], ts[f] * w);
}

// ---------------------------------------------------------------------------
// self-loop + bias + relu; optional BN statistics (sum, sumsq per feature)
// block = 256 = 8 node-rows x 32 features
__global__ void post_kernel(const float* __restrict__ acc, const float* __restrict__ tmp,
                            const float* __restrict__ dinv, const float* __restrict__ bias,
                            float* __restrict__ hout, float* __restrict__ stats,
                            int n, int do_stats) {
    int f = threadIdx.x & 31;
    int r = threadIdx.x >> 5;
    float sum = 0.f, sq = 0.f;
    for (int i = blockIdx.x * 8 + r; i < n; i += gridDim.x * 8) {
        float di = dinv[i];
        float v = acc[(size_t)i * HDIM + f] + tmp[(size_t)i * HDIM + f] * di * di + bias[f];
        v = fmaxf(v, 0.f);
        hout[(size_t)i * HDIM + f] = v;
        sum += v; sq += v * v;
    }
    if (do_stats) {
        __shared__ float ssum[256], ssq[256];
        ssum[threadIdx.x] = sum; ssq[threadIdx.x] = sq;
        __syncthreads();
        if (r == 0) {
#pragma unroll
            for (int k = 1; k < 8; ++k) { sum += ssum[k * 32 + f]; sq += ssq[k * 32 + f]; }
            atomicAdd(&stats[f], sum);
            atomicAdd(&stats[HDIM + f], sq);
        }
    }
}

__global__ void bn_apply(float* __restrict__ hm, const float* __restrict__ stats,
                         const float* __restrict__ g, const float* __restrict__ be, int n) {
    int idx = blockIdx.x * blockDim.x + threadIdx.x;
    if (idx >= n * HDIM) return;
    int f = idx & 31;
    float inv_n = 1.0f / (float)n;
    float mu = stats[f] * inv_n;
    float var = stats[HDIM + f] * inv_n - mu * mu;
    hm[idx] = (hm[idx] - mu) * rsqrtf(var + BN_EPS) * g[f] + be[f];
}

// ---------------------------------------------------------------------------
// ordered compaction of node_type==0 (single block, Hillis-Steele scan)
__global__ void compact_kernel(const int* __restrict__ node_type,
                               int* __restrict__ cand_list, int* __restrict__ cnt, int n) {
    __shared__ int sh[1024];
    __shared__ int base;
    if (threadIdx.x == 0) base = 0;
    __syncthreads();
    for (int start = 0; start < n; start += 1024) {
        int i = start + threadIdx.x;
        int m = (i < n && node_type[i] == 0) ? 1 : 0;
        sh[threadIdx.x] = m;
        __syncthreads();
        for (int off = 1; off < 1024; off <<= 1) {
            int v = (threadIdx.x >= (unsigned)off) ? sh[threadIdx.x - off] : 0;
            __syncthreads();
            sh[threadIdx.x] += v;
            __syncthreads();
        }
        if (m) cand_list[base + sh[threadIdx.x] - 1] = i;
        __syncthreads();
        if (threadIdx.x == 0) base += sh[1023];
        __syncthreads();
    }
    if (threadIdx.x == 0) *cnt = base;
}

// candidate head: relu(emb@Wc1+bc1)@Wc2+bc2 -> out[t]
__global__ void cand_head(const float* __restrict__ hm, const int* __restrict__ cand_list,
                          const int* __restrict__ cnt,
                          const float* __restrict__ Wc1, const float* __restrict__ bc1,
                          const float* __restrict__ Wc2, const float* __restrict__ bc2,
                          float* __restrict__ out) {
    int t = blockIdx.x * blockDim.x + threadIdx.x;
    if (t >= *cnt) return;
    const float* e = hm + (size_t)cand_list[t] * HDIM;
    float score = bc2[0];
#pragma unroll
    for (int j = 0; j < 16; ++j) {
        float a = bc1[j];
#pragma unroll
        for (int k = 0; k < HDIM; ++k) a = fmaf(e[k], Wc1[k * 16 + j], a);
        score = fmaf(fmaxf(a, 0.f), Wc2[j], score);
    }
    out[t] = score;
}

// ---------------------------------------------------------------------------
// softmax pooling: pass 1 = max of row sums (rows are post-relu => >= 0,
// so int-bit atomicMax on non-negative floats is order-preserving)
__global__ void rowsum_max(const float* __restrict__ hm, int* __restrict__ smax_bits, int n) {
    int f = threadIdx.x & 31;
    float local = 0.f;
    for (int i = blockIdx.x * 8 + (threadIdx.x >> 5); i < n; i += gridDim.x * 8) {
        float s = hm[(size_t)i * HDIM + f];
#pragma unroll
        for (int off = 16; off; off >>= 1) s += __shfl_xor(s, off, 32);
        local = fmaxf(local, s);
    }
    if ((threadIdx.x & 31) == 0) atomicMax(smax_bits, __float_as_int(local));
}

// pass 2: Z = sum exp(s - smax), gemb[f] = sum h[i,f]*exp(s_i - smax)
__global__ void softmax_accum(const float* __restrict__ hm, const int* __restrict__ smax_bits,
                              float* __restrict__ z, float* __restrict__ gemb, int n) {
    int f = threadIdx.x & 31;
    float smax = __int_as_float(*smax_bits);
    float gl = 0.f, zl = 0.f;
    for (int i = blockIdx.x * 8 + (threadIdx.x >> 5); i < n; i += gridDim.x * 8) {
        float v = hm[(size_t)i * HDIM + f];
        float s = v;
#pragma unroll
        for (int off = 16; off; off >>= 1) s += __shfl_xor(s, off, 32);
        float w = __expf(s - smax);
        gl += v * w;
        zl += w;
    }
    atomicAdd(&gemb[f], gl);
    if ((threadIdx.x & 31) == 0) atomicAdd(z, zl);
}

// dense head + write dn at out[C] and cand_idx (as float) at out[C+1 ..]
__global__ void final_kernel(const float* __restrict__ gemb, const float* __restrict__ z,
                             const int* __restrict__ cnt,
                             const float* __restrict__ Wd1, const float* __restrict__ bd1,
                             const float* __restrict__ Wd2, const float* __restrict__ bd2,
                             const int* __restrict__ cand_list, float* __restrict__ out) {
    int C = *cnt;
    int t = blockIdx.x * blockDim.x + threadIdx.x;
    if (t == 0) {
        float invz = 1.0f / *z;
        float dn = bd2[0];
        for (int j = 0; j < 16; ++j) {
            float a = bd1[j];
            for (int k = 0; k < HDIM; ++k) a = fmaf(gemb[k] * invz, Wd1[k * 16 + j], a);
            dn = fmaf(fmaxf(a, 0.f), Wd2[j], dn);
        }
        out[C] = dn;
    }
    if (t < C) out[C + 1 + t] = (float)cand_list[t];
}

// ---------------------------------------------------------------------------
extern "C" void kernel_launch(void* const* d_in, const int* in_sizes, int n_in,
                              void* d_out, int out_size, void* d_ws, size_t ws_size,
                              hipStream_t stream) {
    const float* x   = (const float*)d_in[0];
    const int*   ei  = (const int*)  d_in[1];
    const int*   nty = (const int*)  d_in[2];
    const float* W1  = (const float*)d_in[3];  const float* b1  = (const float*)d_in[4];
    const float* W2  = (const float*)d_in[5];  const float* b2  = (const float*)d_in[6];
    const float* W3  = (const float*)d_in[7];  const float* b3  = (const float*)d_in[8];
    const float* g1  = (const float*)d_in[9];  const float* be1 = (const float*)d_in[10];
    const float* g2  = (const float*)d_in[11]; const float* be2 = (const float*)d_in[12];
    const float* Wc1 = (const float*)d_in[13]; const float* bc1 = (const float*)d_in[14];
    const float* Wc2 = (const float*)d_in[15]; const float* bc2 = (const float*)d_in[16];
    const float* Wd1 = (const float*)d_in[17]; const float* bd1 = (const float*)d_in[18];
    const float* Wd2 = (const float*)d_in[19]; const float* bd2 = (const float*)d_in[20];

    const int N = in_sizes[2];
    const int E = in_sizes[1] / 2;
    float* out = (float*)d_out;

    // workspace layout
    char* ws = (char*)d_ws;
    float* dinv      = (float*)ws;                      ws += (size_t)N * sizeof(float);
    float* tmp       = (float*)ws;                      ws += (size_t)N * HDIM * sizeof(float);
    float* acc       = (float*)ws;                      ws += (size_t)N * HDIM * sizeof(float);
    float* hbuf      = (float*)ws;                      ws += (size_t)N * HDIM * sizeof(float);
    float* stats     = (float*)ws;                      ws += 64 * sizeof(float);   // sum,sq
    float* gemb      = (float*)ws;                      ws += 32 * sizeof(float);
    int*   smax_bits = (int*)ws;                        ws += sizeof(int);
    float* zacc      = (float*)ws;                      ws += sizeof(float);
    int*   cnt       = (int*)ws;                        ws += sizeof(int);
    int*   cand_list = (int*)ws;                        ws += (size_t)N * sizeof(int);

    const int TB = 256;
    const int nodeBlocks = (N + TB - 1) / TB;
    const int edgeBlocks = (E + TB - 1) / TB;
    const int elemBlocks = ((size_t)N * HDIM + TB - 1) / TB;
    const int redBlocks  = 768;                 // reduction kernels (8 rows/blk)
    const int mTiles = (N + 15) / 16;
    const int waves  = mTiles * 2;              // 2 col-tiles of 16
    const int wmmaBlocks = (waves + 7) / 8;     // 8 waves / 256-thread block

    // degrees -> dinv
    hipMemsetAsync(dinv, 0, (size_t)N * sizeof(float), stream);
    deg_kernel<<<edgeBlocks, TB, 0, stream>>>(ei, dinv, E);
    dinv_kernel<<<nodeBlocks, TB, 0, stream>>>(dinv, N);

    // ----- conv1 (K=3 scalar GEMM) + BN1
    gemm_k3<<<nodeBlocks, TB, 0, stream>>>(x, W1, tmp, N);
    hipMemsetAsync(acc, 0, (size_t)N * HDIM * sizeof(float), stream);
    hipMemsetAsync(stats, 0, 64 * sizeof(float), stream);
    scatter_edges<<<edgeBlocks, TB, 0, stream>>>(ei, tmp, dinv, acc, E);
    post_kernel<<<redBlocks, TB, 0, stream>>>(acc, tmp, dinv, b1, hbuf, stats, N, 1);
    bn_apply<<<elemBlocks, TB, 0, stream>>>(hbuf, stats, g1, be1, N);

    // ----- conv2 (WMMA GEMM) + BN2
    gemm32_wmma<<<wmmaBlocks, TB, 0, stream>>>(hbuf, W2, tmp, N);
    hipMemsetAsync(acc, 0, (size_t)N * HDIM * sizeof(float), stream);
    hipMemsetAsync(stats, 0, 64 * sizeof(float), stream);
    scatter_edges<<<edgeBlocks, TB, 0, stream>>>(ei, tmp, dinv, acc, E);
    post_kernel<<<redBlocks, TB, 0, stream>>>(acc, tmp, dinv, b2, hbuf, stats, N, 1);
    bn_apply<<<elemBlocks, TB, 0, stream>>>(hbuf, stats, g2, be2, N);

    // ----- conv3 (WMMA GEMM), relu only
    gemm32_wmma<<<wmmaBlocks, TB, 0, stream>>>(hbuf, W3, tmp, N);
    hipMemsetAsync(acc, 0, (size_t)N * HDIM * sizeof(float), stream);
    scatter_edges<<<edgeBlocks, TB, 0, stream>>>(ei, tmp, dinv, acc, E);
    post_kernel<<<redBlocks, TB, 0, stream>>>(acc, tmp, dinv, b3, hbuf, stats, N, 0);

    // ----- candidate compaction + scores
    compact_kernel<<<1, 1024, 0, stream>>>(nty, cand_list, cnt, N);
    cand_head<<<nodeBlocks, TB, 0, stream>>>(hbuf, cand_list, cnt, Wc1, bc1, Wc2, bc2, out);

    // ----- softmax pooling + dense head + index writeout
    hipMemsetAsync(gemb, 0, 32 * sizeof(float), stream);
    hipMemsetAsync(smax_bits, 0, sizeof(int), stream);
    hipMemsetAsync(zacc, 0, sizeof(float), stream);
    rowsum_max<<<redBlocks, TB, 0, stream>>>(hbuf, smax_bits, N);
    softmax_accum<<<redBlocks, TB, 0, stream>>>(hbuf, smax_bits, zacc, gemb, N);
    final_kernel<<<nodeBlocks, TB, 0, stream>>>(gemb, zacc, cnt, Wd1, bd1, Wd2, bd2,
                                                cand_list, out);
}